// MResConv_12171937316913
// MI455X (gfx1250) — compile-verified
//
#include <hip/hip_runtime.h>
#include <cstdint>
#include <cstddef>

typedef float v2f __attribute__((ext_vector_type(2)));
typedef float v8f __attribute__((ext_vector_type(8)));

#define NT 64        // edges per workgroup tile (4 x 16 per wave)
#define MT 128       // output channels per workgroup tile (8 waves x 16)
#define CC 8         // input channels per K-chunk
#define KC (CC * 5)  // K' values per chunk = 40
#define KPAD 44      // padded K' stride (floats): col*44 mod 64 covers all 16
                     // multiples of 4 -> conflict-free ds_load_b64 fragments

// One kernel implements both mesh_conv stages:
//  y[b,o,e] = sum_{c,k} f[b,c,e,k] * W[o,c,k],
//  f = [x, f1+f3, f2+f4, |f1-f3|, |f2-f4|]
// applyBN: input element v -> max(v,0)*bn_s[c] + bn_t[c]  (fused bn1(relu(.)))
// resid != null: out = relu(acc + resid)                  (fused residual)
__global__ __launch_bounds__(256)
void meshconv_wmma(const float* __restrict__ X,
                   const int*   __restrict__ G,
                   const float* __restrict__ W,
                   const float* __restrict__ bn_s,
                   const float* __restrict__ bn_t,
                   const float* __restrict__ resid,
                   float* __restrict__ Y,
                   int Cin, int O, int E, int applyBN)
{
    __shared__ float F[NT * KPAD];     // [edge][K'] transposed feature tile
    __shared__ int   nbr[NT * 4];

    const int tid    = threadIdx.x;
    const int b      = blockIdx.z;
    const int mblock = blockIdx.y;
    const int ebase  = blockIdx.x * NT;

    // Stage the 4 neighbor indices for this 64-edge tile (256 threads = 64x4).
    {
        int el = tid >> 2, j = tid & 3;
        int e = ebase + el; if (e >= E) e = E - 1;
        nbr[el * 4 + j] = G[((size_t)b * E + e) * 4 + j];
    }

    const int lane  = tid & 31;
    const int wave  = tid >> 5;            // 0..7
    const int obase = mblock * MT + wave * 16;
    const int hi    = (lane >= 16) ? 1 : 0;
    const int l15   = lane & 15;

    v8f acc0 = {0.f, 0.f, 0.f, 0.f, 0.f, 0.f, 0.f, 0.f};
    v8f acc1 = acc0, acc2 = acc0, acc3 = acc0;

    const int Ktot    = Cin * 5;
    const float* wrow = W + (size_t)(obase + l15) * Ktot;  // per-lane A row

    const int nchunks = Cin / CC;
    for (int cb = 0; cb < nchunks; ++cb) {
        __syncthreads();   // protect F from previous chunk's readers
        // Build features: 8 channels x 64 edges = 512 (c,e) pairs, 2/thread.
        #pragma unroll
        for (int it = 0; it < 2; ++it) {
            int p  = tid + it * 256;
            int cl = p >> 6;               // 0..7
            int el = p & 63;               // 0..63
            int c  = cb * CC + cl;
            int e  = ebase + el; if (e >= E) e = E - 1;
            const float* xr = X + ((size_t)b * Cin + c) * E;
            float xc = xr[e];
            float f1 = xr[nbr[el * 4 + 0]];
            float f2 = xr[nbr[el * 4 + 1]];
            float f3 = xr[nbr[el * 4 + 2]];
            float f4 = xr[nbr[el * 4 + 3]];
            if (applyBN) {
                float s = bn_s[c], t = bn_t[c];
                xc = fmaxf(xc, 0.f) * s + t;
                f1 = fmaxf(f1, 0.f) * s + t;
                f2 = fmaxf(f2, 0.f) * s + t;
                f3 = fmaxf(f3, 0.f) * s + t;
                f4 = fmaxf(f4, 0.f) * s + t;
            }
            float* fe = F + el * KPAD + cl * 5;
            fe[0] = xc;
            fe[1] = f1 + f3;
            fe[2] = f2 + f4;
            fe[3] = fabsf(f1 - f3);
            fe[4] = fabsf(f2 - f4);
        }
        __syncthreads();

        const float* wp = wrow + cb * KC;  // chunk base in W row
        #pragma unroll
        for (int ks = 0; ks < KC / 4; ++ks) {
            // A 16x4 f32: lanes 0-15 hold (K,K+1) for M=lane; 16-31 hold (K+2,K+3)
            v2f a = *reinterpret_cast<const v2f*>(wp + ks * 4 + 2 * hi);
            int krow = ks * 4 + 2 * hi;    // even -> 8B-aligned LDS loads
            v2f b0 = *reinterpret_cast<const v2f*>(F + (l15     ) * KPAD + krow);
            v2f b1 = *reinterpret_cast<const v2f*>(F + (l15 + 16) * KPAD + krow);
            v2f b2 = *reinterpret_cast<const v2f*>(F + (l15 + 32) * KPAD + krow);
            v2f b3 = *reinterpret_cast<const v2f*>(F + (l15 + 48) * KPAD + krow);
            acc0 = __builtin_amdgcn_wmma_f32_16x16x4_f32(
                false, a, false, b0, (short)0, acc0, false, false);
            acc1 = __builtin_amdgcn_wmma_f32_16x16x4_f32(
                false, a, false, b1, (short)0, acc1, false, false);
            acc2 = __builtin_amdgcn_wmma_f32_16x16x4_f32(
                false, a, false, b2, (short)0, acc2, false, false);
            acc3 = __builtin_amdgcn_wmma_f32_16x16x4_f32(
                false, a, false, b3, (short)0, acc3, false, false);
        }
    }

    // Epilogue. C/D layout: VGPR r -> M = r + 8*hi, N = lane&15.
    int e0 = ebase + l15;
    int e1 = e0 + 16, e2 = e0 + 32, e3 = e0 + 48;
    #pragma unroll
    for (int r = 0; r < 8; ++r) {
        int o = obase + r + 8 * hi;
        size_t row = ((size_t)b * O + o) * E;
        if (resid) {
            if (e0 < E) { float v = acc0[r] + resid[row + e0]; Y[row + e0] = fmaxf(v, 0.f); }
            if (e1 < E) { float v = acc1[r] + resid[row + e1]; Y[row + e1] = fmaxf(v, 0.f); }
            if (e2 < E) { float v = acc2[r] + resid[row + e2]; Y[row + e2] = fmaxf(v, 0.f); }
            if (e3 < E) { float v = acc3[r] + resid[row + e3]; Y[row + e3] = fmaxf(v, 0.f); }
        } else {
            if (e0 < E) Y[row + e0] = acc0[r];
            if (e1 < E) Y[row + e1] = acc1[r];
            if (e2 < E) Y[row + e2] = acc2[r];
            if (e3 < E) Y[row + e3] = acc3[r];
        }
    }
}

// Per-channel BN training stats over relu(y0), folded to affine s,t.
__global__ __launch_bounds__(256)
void bn_stats(const float* __restrict__ Y0,
              const float* __restrict__ gamma,
              const float* __restrict__ beta,
              float* __restrict__ bn_s,
              float* __restrict__ bn_t,
              int B, int C, int E)
{
    __shared__ float ssum[256];
    __shared__ float ssq[256];
    const int c = blockIdx.x;
    float sum = 0.f, sq = 0.f;
    for (int b = 0; b < B; ++b) {
        const float* row = Y0 + ((size_t)b * C + c) * E;
        for (int e = threadIdx.x; e < E; e += 256) {
            float v = fmaxf(row[e], 0.f);
            sum += v; sq += v * v;
        }
    }
    ssum[threadIdx.x] = sum;
    ssq[threadIdx.x]  = sq;
    __syncthreads();
    for (int s = 128; s > 0; s >>= 1) {
        if (threadIdx.x < s) {
            ssum[threadIdx.x] += ssum[threadIdx.x + s];
            ssq[threadIdx.x]  += ssq[threadIdx.x + s];
        }
        __syncthreads();
    }
    if (threadIdx.x == 0) {
        float n    = (float)B * (float)E;
        float mean = ssum[0] / n;
        float var  = ssq[0] / n - mean * mean;
        float rs   = rsqrtf(var + 1e-5f);
        float g    = gamma[c];
        bn_s[c] = g * rs;
        bn_t[c] = beta[c] - mean * g * rs;
    }
}

extern "C" void kernel_launch(void* const* d_in, const int* in_sizes, int n_in,
                              void* d_out, int out_size, void* d_ws, size_t ws_size,
                              hipStream_t stream)
{
    (void)in_sizes; (void)n_in; (void)out_size; (void)ws_size;
    const float* x     = (const float*)d_in[0];
    const int*   gemm  = (const int*)  d_in[1];
    const float* W0    = (const float*)d_in[2];
    const float* gamma = (const float*)d_in[3];
    const float* beta  = (const float*)d_in[4];
    const float* W1    = (const float*)d_in[5];

    const int B = 4, Cin = 128, O = 256, E = 30000;

    float*  y0       = (float*)d_ws;                    // [B, O, E] conv0 out
    size_t  y0_elems = (size_t)B * O * E;
    float*  bn_s     = y0 + y0_elems;                   // [O]
    float*  bn_t     = bn_s + O;                        // [O]
    float*  out      = (float*)d_out;

    dim3 block(256);
    dim3 grid((E + NT - 1) / NT, O / MT, B);            // 469 x 2 x 4

    // conv0: x -> y0
    meshconv_wmma<<<grid, block, 0, stream>>>(
        x, gemm, W0, nullptr, nullptr, nullptr, y0, Cin, O, E, 0);

    // bn1 training stats over relu(y0) -> affine (s, t)
    bn_stats<<<dim3(O), block, 0, stream>>>(y0, gamma, beta, bn_s, bn_t, B, O, E);

    // conv1 with fused bn(relu(.)) on inputs, fused +y0 residual and relu on output
    meshconv_wmma<<<grid, block, 0, stream>>>(
        y0, gemm, W1, bn_s, bn_t, y0, out, O, O, E, 1);
}